// AdditiveAttention_23356032155851
// MI455X (gfx1250) — compile-verified
//
#include <hip/hip_runtime.h>

// ---------------------------------------------------------------------------
// AdditiveAttention on gfx1250 (MI455X), fp32 end-to-end.
//   B=8, K=512, Q=128, DK=DQ=DV=H=256
// Pipeline:
//   1) proj_kernel (x2): qp = queries @ Wq^T, kp = keys @ Wk^T  (WMMA f32 16x16x4)
//   2) fused_kernel: logits = sum_h Ws[h]*tanh(qp+kp) (TRANS pipe),
//      masked softmax over k, then out = atten @ values (WMMA f32 16x16x4).
//      kp / values tiles staged into LDS with the Tensor Data Mover
//      (tensor_load_to_lds, TENSORcnt), double-buffered.
// ---------------------------------------------------------------------------

typedef float v2f __attribute__((ext_vector_type(2)));
typedef float v8f __attribute__((ext_vector_type(8)));
typedef unsigned int v4u __attribute__((ext_vector_type(4)));
typedef int v4i __attribute__((ext_vector_type(4)));
typedef int v8i __attribute__((ext_vector_type(8)));

#define NEGMASK -1000000.0f
#define AT_STRIDE 516  // atten row stride in floats (pad: 516%64==4 -> conflict-free)
#define KT 32          // k rows per staged tile
#define SMEM_FLOATS (2 * KT * 256 + 16 * AT_STRIDE)
#define SMEM_BYTES (SMEM_FLOATS * 4)

#if __has_builtin(__builtin_amdgcn_tensor_load_to_lds) && \
    __has_builtin(__builtin_amdgcn_s_wait_tensorcnt)
#define HAVE_TDM 1
#else
#define HAVE_TDM 0
#endif

__device__ __forceinline__ float fast_tanh(float x) {
#if __has_builtin(__builtin_amdgcn_tanhf)
  return __builtin_amdgcn_tanhf(x);
#elif __has_builtin(__builtin_amdgcn_tanh_f32)
  return __builtin_amdgcn_tanh_f32(x);
#else
  float e = __builtin_amdgcn_exp2f(2.8853900817779268f * x);
  return 1.0f - 2.0f / (e + 1.0f);
#endif
}

__device__ __forceinline__ float fast_exp2(float x) {
#if __has_builtin(__builtin_amdgcn_exp2f)
  return __builtin_amdgcn_exp2f(x);
#else
  return exp2f(x);
#endif
}

__device__ __forceinline__ float fast_rcp(float x) {
#if __has_builtin(__builtin_amdgcn_rcpf)
  return __builtin_amdgcn_rcpf(x);
#else
  return 1.0f / x;
#endif
}

#if HAVE_TDM
// TDM 2D tile load: `rows` rows of 256 f32, row stride 256 f32, into LDS.
// D# per CDNA5 ISA (group0 128b, group1 256b).
__device__ __forceinline__ void tdm_load_rows(unsigned lds_byte_off,
                                              const float* gsrc, unsigned rows) {
  unsigned long long ga = (unsigned long long)(uintptr_t)gsrc;
  const unsigned D0 = 256u;  // elements per row (contiguous dim)
  v4u g0 = {
      1u,                                               // count=1 (valid user D#)
      lds_byte_off,                                     // lds_addr
      (unsigned)ga,                                     // global_addr[31:0]
      (unsigned)((ga >> 32) & 0x01FFFFFFu) | (2u << 30) // addr[56:32] | type=2
  };
  v8i g1 = {
      (int)(2u << 16),                            // wg_mask=0 | data_size=2 (4B)
      (int)((D0 & 0xFFFFu) << 16),                // tensor_dim0 lo16 in [31:16]
      (int)((D0 >> 16) | ((rows & 0xFFFFu) << 16)),  // tdim0 hi | tensor_dim1 lo
      (int)((rows >> 16) | ((D0 & 0xFFFFu) << 16)),  // tdim1 hi | tile_dim0=256
      (int)(rows & 0xFFFFu),                      // tile_dim1=rows | tile_dim2=0
      (int)D0,                                    // tensor_dim0_stride (lo32)
      0,                                          // stride0 hi | stride1 lo
      0                                           // stride1 hi
  };
  v4i gz = {0, 0, 0, 0};
#if __has_include(<hip/amd_detail/amd_gfx1250_TDM.h>)
  v8i gz8 = {0, 0, 0, 0, 0, 0, 0, 0};
  __builtin_amdgcn_tensor_load_to_lds(g0, g1, gz, gz, gz8, 0);
#else
  __builtin_amdgcn_tensor_load_to_lds(g0, g1, gz, gz, 0);
#endif
}
__device__ __forceinline__ unsigned lds_off(const void* p) {
  return (unsigned)(uintptr_t)p;  // low 32 bits of generic ptr = LDS byte offset
}
#endif

// ---------------------------------------------------------------------------
// Kernel 1: P[b][r][h] = sum_d X[b][r][d] * W[h][col0 + d]   (d = 0..255)
// One wave per 16x16 tile, WMMA f32 16x16x4 (layouts per ISA 7.12.2).
// ---------------------------------------------------------------------------
__global__ __launch_bounds__(32) void proj_kernel(
    const float* __restrict__ X, const float* __restrict__ W,
    float* __restrict__ P, int R, int col0) {
  const int b = blockIdx.z, r0 = blockIdx.x * 16, h0 = blockIdx.y * 16;
  const int lane = threadIdx.x, half = lane >> 4, l15 = lane & 15;

  const float* Xrow = X + ((size_t)b * R + (r0 + l15)) * 256;
  const float* Wrow = W + (size_t)(h0 + l15) * 512 + col0;

  v8f c = {};
#pragma unroll 4
  for (int d = 0; d < 256; d += 4) {
    v2f a  = *(const v2f*)(Xrow + d + 2 * half);  // A: K={0,1}/{2,3} per lane half
    v2f bm = *(const v2f*)(Wrow + d + 2 * half);  // B[kk][n] = W[h0+n][col0+d+kk]
    c = __builtin_amdgcn_wmma_f32_16x16x4_f32(false, a, false, bm, (short)0, c,
                                              false, false);
  }
  float* Prow = P + ((size_t)b * R + r0) * 256 + h0 + l15;
#pragma unroll
  for (int i = 0; i < 8; ++i) Prow[(size_t)(i + 8 * half) * 256] = c[i];
}

// ---------------------------------------------------------------------------
// Kernel 2: fused logits(tanh) + masked softmax + atten@values.
// Block = 512 threads (16 waves; wave w = query row w and value-col tile w).
// LDS: 2 x [KT][256] staging buffers + atten[16][AT_STRIDE].
// ---------------------------------------------------------------------------
__global__ __launch_bounds__(512) void fused_kernel(
    const float* __restrict__ qp, const float* __restrict__ kp,
    const float* __restrict__ values, const long long* __restrict__ vlen,
    const float* __restrict__ wscore, float* __restrict__ out) {
  extern __shared__ float smem[];
  float* buf0   = smem;                 // [KT][256]
  float* buf1   = smem + KT * 256;      // [KT][256]
  float* at_lds = smem + 2 * KT * 256;  // [16][AT_STRIDE]

  const int b = blockIdx.y, q0 = blockIdx.x * 16;
  const int tid = threadIdx.x, w = tid >> 5, lane = tid & 31;
  const int q = q0 + w;

  const float* kpb = kp + (size_t)b * 512 * 256;
  const float* vb  = values + (size_t)b * 512 * 256;

#if HAVE_TDM
  if (w == 0) tdm_load_rows(lds_off(buf0), kpb, KT);  // prefetch tile 0
#endif

  // Per-lane h slices: h = lane*4 + j (j<4) and h = 128 + lane*4 + (j-4).
  // Dense LDS bank coverage for the two b128 loads per k.
  float qreg[8], wreg[8];
  {
    const float* qrow = qp + ((size_t)b * 128 + q) * 256;
    float4 a = *(const float4*)(qrow + lane * 4);
    float4 c4 = *(const float4*)(qrow + 128 + lane * 4);
    qreg[0] = a.x; qreg[1] = a.y; qreg[2] = a.z; qreg[3] = a.w;
    qreg[4] = c4.x; qreg[5] = c4.y; qreg[6] = c4.z; qreg[7] = c4.w;
    float4 d = *(const float4*)(wscore + lane * 4);
    float4 e = *(const float4*)(wscore + 128 + lane * 4);
    wreg[0] = d.x; wreg[1] = d.y; wreg[2] = d.z; wreg[3] = d.w;
    wreg[4] = e.x; wreg[5] = e.y; wreg[6] = e.z; wreg[7] = e.w;
  }
  const long long vl = vlen[b];

  // ---- logits: per k, reduce over h (8 regs/lane x 32 lanes) --------------
  for (int t = 0; t < 512 / KT; ++t) {
#if HAVE_TDM
    if (w == 0) {
      if (t < 15) {
        tdm_load_rows(lds_off((t & 1) ? buf0 : buf1), kpb + (t + 1) * KT * 256, KT);
        __builtin_amdgcn_s_wait_tensorcnt(1);  // tile t complete
      } else {
        __builtin_amdgcn_s_wait_tensorcnt(0);
      }
    }
    __syncthreads();
    const float* bufc = (t & 1) ? buf1 : buf0;
#else
    __syncthreads();
    {
      const float4* src = (const float4*)(kpb + (size_t)t * KT * 256);
      float4* dst = (float4*)buf0;
#pragma unroll
      for (int i = 0; i < (KT * 64) / 512; ++i) dst[tid + i * 512] = src[tid + i * 512];
    }
    __syncthreads();
    const float* bufc = buf0;
#endif

    for (int kk = 0; kk < KT; ++kk) {
      const float* krow = bufc + kk * 256;
      float4 k0 = *(const float4*)(krow + lane * 4);
      float4 k1 = *(const float4*)(krow + 128 + lane * 4);
      float p;
      p  = wreg[0] * fast_tanh(qreg[0] + k0.x);
      p += wreg[1] * fast_tanh(qreg[1] + k0.y);
      p += wreg[2] * fast_tanh(qreg[2] + k0.z);
      p += wreg[3] * fast_tanh(qreg[3] + k0.w);
      p += wreg[4] * fast_tanh(qreg[4] + k1.x);
      p += wreg[5] * fast_tanh(qreg[5] + k1.y);
      p += wreg[6] * fast_tanh(qreg[6] + k1.z);
      p += wreg[7] * fast_tanh(qreg[7] + k1.w);
      p += __shfl_xor(p, 1, 32);
      p += __shfl_xor(p, 2, 32);
      p += __shfl_xor(p, 4, 32);
      p += __shfl_xor(p, 8, 32);
      p += __shfl_xor(p, 16, 32);
      if (lane == 0) at_lds[w * AT_STRIDE + t * KT + kk] = p;
    }
#if HAVE_TDM
    __syncthreads();  // reads of buf[t&1] done -> safe DMA target at t+1
#endif
  }

#if HAVE_TDM
  if (w == 0) tdm_load_rows(lds_off(buf0), vb, KT);  // prefetch values tile 0
#endif

  // ---- masked softmax over k (row w, 512 values, 16 per lane) -------------
  {
    float* row = at_lds + w * AT_STRIDE;
    float vals[16];
    const int kbase = lane * 16;
    float lmax = NEGMASK;
#pragma unroll
    for (int j = 0; j < 16; ++j) {
      int k = kbase + j;
      float v = row[k];
      v = ((long long)k < vl) ? v : NEGMASK;
      vals[j] = v;
      lmax = fmaxf(lmax, v);
    }
    lmax = fmaxf(lmax, __shfl_xor(lmax, 1, 32));
    lmax = fmaxf(lmax, __shfl_xor(lmax, 2, 32));
    lmax = fmaxf(lmax, __shfl_xor(lmax, 4, 32));
    lmax = fmaxf(lmax, __shfl_xor(lmax, 8, 32));
    lmax = fmaxf(lmax, __shfl_xor(lmax, 16, 32));
    float lsum = 0.0f;
#pragma unroll
    for (int j = 0; j < 16; ++j) {
      float e = fast_exp2((vals[j] - lmax) * 1.44269504088896340736f);
      vals[j] = e;
      lsum += e;
    }
    lsum += __shfl_xor(lsum, 1, 32);
    lsum += __shfl_xor(lsum, 2, 32);
    lsum += __shfl_xor(lsum, 4, 32);
    lsum += __shfl_xor(lsum, 8, 32);
    lsum += __shfl_xor(lsum, 16, 32);
    float inv = fast_rcp(lsum);
#pragma unroll
    for (int j = 0; j < 16; ++j) row[kbase + j] = vals[j] * inv;
  }
  __syncthreads();

  // ---- out[b][q0+m][n0+n] = sum_k atten[m][k] * values[b][k][n0+n] --------
  {
    const int half = lane >> 4, l15 = lane & 15;
    const int n0 = w * 16;
    v8f cacc = {};
    for (int kt = 0; kt < 512 / KT; ++kt) {
#if HAVE_TDM
      if (w == 0) {
        if (kt < 15) {
          tdm_load_rows(lds_off((kt & 1) ? buf0 : buf1), vb + (kt + 1) * KT * 256, KT);
          __builtin_amdgcn_s_wait_tensorcnt(1);
        } else {
          __builtin_amdgcn_s_wait_tensorcnt(0);
        }
      }
      __syncthreads();
      const float* bufc = (kt & 1) ? buf1 : buf0;
#else
      __syncthreads();
      {
        const float4* src = (const float4*)(vb + (size_t)kt * KT * 256);
        float4* dst = (float4*)buf0;
#pragma unroll
        for (int i = 0; i < (KT * 64) / 512; ++i) dst[tid + i * 512] = src[tid + i * 512];
      }
      __syncthreads();
      const float* bufc = buf0;
#endif
#pragma unroll 4
      for (int kk2 = 0; kk2 < KT; kk2 += 4) {
        // A: atten rows, row=l15, K = kt*KT + kk2 + {0,1}/{2,3}
        v2f a = *(const v2f*)(at_lds + l15 * AT_STRIDE + kt * KT + kk2 + 2 * half);
        // B[r][n] = staged values[kk2+r][n0+n]
        v2f bm;
        bm.x = bufc[(kk2 + 2 * half) * 256 + n0 + l15];
        bm.y = bufc[(kk2 + 2 * half + 1) * 256 + n0 + l15];
        cacc = __builtin_amdgcn_wmma_f32_16x16x4_f32(false, a, false, bm, (short)0,
                                                     cacc, false, false);
      }
#if HAVE_TDM
      __syncthreads();
#endif
    }
    float* orow = out + ((size_t)b * 128 + q0) * 256 + n0 + l15;
#pragma unroll
    for (int i = 0; i < 8; ++i) orow[(size_t)(i + 8 * half) * 256] = cacc[i];
  }
}

// ---------------------------------------------------------------------------
extern "C" void kernel_launch(void* const* d_in, const int* in_sizes, int n_in,
                              void* d_out, int out_size, void* d_ws, size_t ws_size,
                              hipStream_t stream) {
  const float*     keys      = (const float*)d_in[0];  // (8,512,256)
  const float*     values    = (const float*)d_in[1];  // (8,512,256)
  const float*     queries   = (const float*)d_in[2];  // (8,128,256)
  const long long* valid_len = (const long long*)d_in[3];
  const float*     W_hidden  = (const float*)d_in[4];  // (256,512)
  const float*     W_score   = (const float*)d_in[5];  // (256,)
  float*           out       = (float*)d_out;          // (8,128,256)

  char*  ws  = (char*)d_ws;
  float* qpw = (float*)ws;                               // 1 MB
  float* kpw = (float*)(ws + (size_t)8 * 128 * 256 * 4); // 4 MB

  (void)hipFuncSetAttribute(reinterpret_cast<const void*>(&fused_kernel),
                            hipFuncAttributeMaxDynamicSharedMemorySize, SMEM_BYTES);

  proj_kernel<<<dim3(128 / 16, 256 / 16, 8), 32, 0, stream>>>(queries, W_hidden,
                                                              qpw, 128, 0);
  proj_kernel<<<dim3(512 / 16, 256 / 16, 8), 32, 0, stream>>>(keys, W_hidden,
                                                              kpw, 512, 256);
  fused_kernel<<<dim3(128 / 16, 8), 512, SMEM_BYTES, stream>>>(qpw, kpw, values,
                                                               valid_len, W_score,
                                                               out);
}